// OccGridAccelBatchedDynamic_Ema_21242908246601
// MI455X (gfx1250) — compile-verified
//
#include <hip/hip_runtime.h>

// Occupancy-grid gather (OccGridAccelBatchedDynamic EMA lookup), MI455X / gfx1250.
//
// Pure memory-bound irregular gather: ~48 MB coalesced streams + ~256 MB of
// unavoidable random cacheline fills from the 256 MB occ grid.  No matrix
// FLOPs -> no WMMA.  Strategy: 4 points per thread (b128 coalesced streams,
// 4 gathers in flight per thread), non-temporal TH hints on every access
// (read-once data, no-reuse gather; keep the 192 MB L2 from thrashing),
// keyframe table in SGPRs via uniform loads, branchless nearest-keyframe.

#define KF_N 16  // NUM_FRAMES in the reference

// Native clang vectors: required by __builtin_nontemporal_load/store
// (HIP's float4 is a struct and is rejected by the builtin).
typedef float vf4 __attribute__((ext_vector_type(4)));
typedef int   vi4 __attribute__((ext_vector_type(4)));

__device__ __forceinline__ int nearest_kf(const float* __restrict__ k, float tv) {
    // jnp.searchsorted(left) == count of keys strictly < tv, clipped to [1, F-1];
    // then pick the closer of keys[i-1], keys[i].  Fully unrolled, branchless
    // (v_cmp + v_cndmask), no runtime-indexed register array -> no scratch.
    float L = k[0];
    float R = k[1];
    int   idx = 1;
#pragma unroll
    for (int i = 1; i < KF_N - 1; ++i) {
        bool lt = (k[i] < tv);
        idx = lt ? (i + 1) : idx;
        L   = lt ? k[i]     : L;
        R   = lt ? k[i + 1] : R;
    }
    return (tv - L <= R - tv) ? (idx - 1) : idx;
}

__device__ __forceinline__ int voxel(float p, float resf, int resm1) {
    int g = (int)floorf((p * 0.5f + 0.5f) * resf);
    g = (g < 0) ? 0 : g;
    g = (g > resm1) ? resm1 : g;
    return g;
}

__global__ __launch_bounds__(256) void occ_gather_vec4(
    const vf4* __restrict__ pts4,   // 3 x vf4 per quad of points (AoS [N,3])
    const vi4* __restrict__ bidx4,
    const vf4* __restrict__ ts4,
    const float* __restrict__ occ,
    const float* __restrict__ kfp,  // 16 sorted keyframe times
    vf4*       __restrict__ out4,
    int nq, int res)
{
    int q = blockIdx.x * blockDim.x + threadIdx.x;
    if (q >= nq) return;

    // Keyframes: uniform address + constant index -> scalar loads into SGPRs.
    float kf[KF_N];
#pragma unroll
    for (int i = 0; i < KF_N; ++i) kf[i] = kfp[i];

    // Coalesced b128 non-temporal streams (read-once).
    vf4 p0 = __builtin_nontemporal_load(&pts4[3 * q + 0]);
    vf4 p1 = __builtin_nontemporal_load(&pts4[3 * q + 1]);
    vf4 p2 = __builtin_nontemporal_load(&pts4[3 * q + 2]);
    vf4 tq = __builtin_nontemporal_load(&ts4[q]);
    vi4 bq = __builtin_nontemporal_load(&bidx4[q]);

    // De-swizzle AoS [x0 y0 z0 x1 | y1 z1 x2 y2 | z2 x3 y3 z3]
    float px[4] = {p0.x, p0.w, p1.z, p2.y};
    float py[4] = {p0.y, p1.x, p1.w, p2.z};
    float pz[4] = {p0.z, p1.y, p2.x, p2.w};
    float tv[4] = {tq.x, tq.y, tq.z, tq.w};
    int   bv[4] = {bq.x, bq.y, bq.z, bq.w};

    const float    resf  = (float)res;
    const int      resm1 = res - 1;
    const unsigned res2  = (unsigned)(res * res);
    const unsigned res3  = res2 * (unsigned)res;

    float r[4];
#pragma unroll
    for (int j = 0; j < 4; ++j) {
        int gx = voxel(px[j], resf, resm1);
        int gy = voxel(py[j], resf, resm1);
        int gz = voxel(pz[j], resf, resm1);
        int fi = nearest_kf(kf, tv[j]);
        unsigned flat = ((unsigned)bv[j] * (unsigned)KF_N + (unsigned)fi) * res3
                      + (unsigned)gx * res2 + (unsigned)gy * (unsigned)res + (unsigned)gz;
        // Random 4B gather, no reuse -> TH_LOAD_NT so we don't pollute L2.
        r[j] = __builtin_nontemporal_load(&occ[flat]);
    }

    vf4 o = {r[0], r[1], r[2], r[3]};
    __builtin_nontemporal_store(o, &out4[q]);
}

// Scalar tail for N % 4 points (N = 2,097,152 is divisible by 4, but stay safe).
__global__ __launch_bounds__(64) void occ_gather_tail(
    const float* __restrict__ pts,
    const int*   __restrict__ bidx,
    const float* __restrict__ ts,
    const float* __restrict__ occ,
    const float* __restrict__ kfp,
    float*       __restrict__ out,
    int base, int n, int res)
{
    int i = base + blockIdx.x * blockDim.x + threadIdx.x;
    if (i >= n) return;

    float kf[KF_N];
#pragma unroll
    for (int k = 0; k < KF_N; ++k) kf[k] = kfp[k];

    const float    resf  = (float)res;
    const int      resm1 = res - 1;
    const unsigned res2  = (unsigned)(res * res);
    const unsigned res3  = res2 * (unsigned)res;

    int gx = voxel(pts[3 * i + 0], resf, resm1);
    int gy = voxel(pts[3 * i + 1], resf, resm1);
    int gz = voxel(pts[3 * i + 2], resf, resm1);
    int fi = nearest_kf(kf, ts[i]);
    unsigned flat = ((unsigned)bidx[i] * (unsigned)KF_N + (unsigned)fi) * res3
                  + (unsigned)gx * res2 + (unsigned)gy * (unsigned)res + (unsigned)gz;
    out[i] = __builtin_nontemporal_load(&occ[flat]);
}

extern "C" void kernel_launch(void* const* d_in, const int* in_sizes, int n_in,
                              void* d_out, int out_size, void* d_ws, size_t ws_size,
                              hipStream_t stream) {
    const float* pts  = (const float*)d_in[0];   // [N,3] f32
    const int*   bidx = (const int*)d_in[1];     // [N] int
    const float* ts   = (const float*)d_in[2];   // [N] f32
    const float* occ  = (const float*)d_in[3];   // [B*F, R,R,R] f32
    const float* kfp  = (const float*)d_in[4];   // [F] f32 sorted
    // d_in[5] = num_frames scalar on device; F == in_sizes[4] == 16 (KF_N).

    const int N   = in_sizes[2];                 // number of points
    const int res = 64;                          // RES in the reference

    float* out = (float*)d_out;

    const int nq = N >> 2;                       // quads of points
    if (nq > 0) {
        const int threads = 256;
        const int blocks  = (nq + threads - 1) / threads;
        occ_gather_vec4<<<blocks, threads, 0, stream>>>(
            (const vf4*)pts, (const vi4*)bidx, (const vf4*)ts,
            occ, kfp, (vf4*)out, nq, res);
    }
    const int tail = N - (nq << 2);
    if (tail > 0) {
        occ_gather_tail<<<1, 64, 0, stream>>>(
            pts, bidx, ts, occ, kfp, out, nq << 2, N, res);
    }
}